// VGAEEncoder_36867999269274
// MI455X (gfx1250) — compile-verified
//
#include <hip/hip_runtime.h>

typedef float v2f __attribute__((ext_vector_type(2)));
typedef float v8f __attribute__((ext_vector_type(8)));

#define HD 256          // hidden width H
#define EMBD 128        // embedding width
#define NEG_HUGE (-3.402823466e38f)

// ---------------------------------------------------------------------------
// fp32 WMMA GEMM:  C[M,256] = act( A[M,K] @ W[K,256] + bias )
// Requirements: M % 16 == 0, Nc == 256, K % 4 == 0 (all hold here).
// Block = 256 threads = 8 waves; block tile = 16 rows x 256 cols,
// each wave owns a 16x32 tile (2 accumulators share one A fragment) via
// V_WMMA_F32_16X16X4_F32.  A tile (16 x K) staged into LDS with CDNA5
// async global->LDS copies (ASYNCcnt); +4 float row pad keeps the b128
// copies 16B-aligned while making ds_load_b64 reads bank-conflict-free
// (row stride mod 64 banks == 4 -> lanes 0-15 hit banks 4*l).
// ---------------------------------------------------------------------------
__global__ __launch_bounds__(256) void gemm_wmma_f32(
    const float* __restrict__ A, const float* __restrict__ W,
    const float* __restrict__ bias, float* __restrict__ C,
    int M, int K, int Nc, int relu)
{
    extern __shared__ float lds[];          // 16 * (K + 4) floats
    const int KP   = K + 4;
    const int tid  = threadIdx.x;
    const int wave = tid >> 5;
    const int lane = tid & 31;
    const int l16  = lane & 15;
    const int half = lane >> 4;             // 0: lanes 0-15, 1: lanes 16-31

    const int rowBase = blockIdx.x * 16;
    const int col     = wave * 32 + l16;    // first 16-col tile
    const int col2    = col + 16;           // second 16-col tile

    // ---- async stage of the 16-row A tile (16B granules, ASYNCcnt) ----
    {
        const float* Ablk = A + (size_t)rowBase * K;
        const int kq = K >> 2;              // b128 transfers per row
        const int nt = 16 * kq;             // total transfers
        for (int j = tid; j < nt; j += 256) {
            const int r  = j / kq;
            const int c4 = (j - r * kq) << 2;          // float index in row
            const unsigned ldsoff = (unsigned)((r * KP + c4) * 4); // bytes
            const float* gptr = Ablk + (size_t)r * K + c4;
            asm volatile("global_load_async_to_lds_b128 %0, %1, off"
                         :: "v"(ldsoff), "v"(gptr) : "memory");
        }
        asm volatile("s_wait_asynccnt 0x0" ::: "memory");
    }
    __syncthreads();

    // ---- K loop: 1 shared A fragment, 2 B fragments, 2 WMMAs per step ----
    v8f acc0 = {};
    v8f acc1 = {};
    const float* ap = lds + l16 * KP + half * 2;            // A frag ptr
    const float* wp = W + (size_t)(half * 2) * Nc + col;    // B frag ptr
    for (int k = 0; k < K; k += 4) {
        v2f a;                              // A 16x4: half selects K{0,1}/{2,3}
        a.x = ap[0];
        a.y = ap[1];
        v2f b0, b1;                         // B 4x16: VGPR v = rows v / v+2
        b0.x = wp[0];
        b0.y = wp[Nc];
        b1.x = wp[16];
        b1.y = wp[Nc + 16];
        acc0 = __builtin_amdgcn_wmma_f32_16x16x4_f32(
            false, a, false, b0, (short)0, acc0, false, false);
        acc1 = __builtin_amdgcn_wmma_f32_16x16x4_f32(
            false, a, false, b1, (short)0, acc1, false, false);
        ap += 4;
        wp += (size_t)4 * Nc;
    }

    // ---- epilogue: bias + relu + store (C layout: lanes 16-31 hold M+8) ----
    const float bv0 = bias ? bias[col]  : 0.0f;
    const float bv1 = bias ? bias[col2] : 0.0f;
    const int r0 = rowBase + half * 8;
    #pragma unroll
    for (int v = 0; v < 8; ++v) {
        float x0 = acc0[v] + bv0;
        float x1 = acc1[v] + bv1;
        if (relu) { x0 = fmaxf(x0, 0.0f); x1 = fmaxf(x1, 0.0f); }
        C[(size_t)(r0 + v) * Nc + col]  = x0;
        C[(size_t)(r0 + v) * Nc + col2] = x1;
    }
}

// ---------------------------------------------------------------------------
// Helpers
// ---------------------------------------------------------------------------
__global__ void zero_f32(float* __restrict__ p, long long n)
{
    long long i = (long long)blockIdx.x * blockDim.x + threadIdx.x;
    if (i < n) p[i] = 0.0f;
}

// degree count at dst (self-loop +1 folded into rsqrt below)
__global__ void deg_count(const int* __restrict__ ei, float* __restrict__ deg,
                          long long E)
{
    long long e = (long long)blockIdx.x * blockDim.x + threadIdx.x;
    if (e < E) atomicAdd(&deg[ei[(size_t)E + e]], 1.0f);
}

__global__ void deg_to_dinv(float* __restrict__ d, int N)
{
    int i = blockIdx.x * blockDim.x + threadIdx.x;
    if (i < N) d[i] = rsqrtf(d[i] + 1.0f);   // +1 self-loop; always > 0
}

// scatter-add over edges: agg[dst] += m[src] * dinv[src]*dinv[dst]
// one thread = one edge x one 4-float chunk (HD/4 = 64 chunks per edge)
__global__ void scatter_edges(const int* __restrict__ ei,
                              const float* __restrict__ m,
                              const float* __restrict__ dinv,
                              float* __restrict__ agg, long long E)
{
    long long idx = (long long)blockIdx.x * blockDim.x + threadIdx.x;
    long long total = E * (HD / 4);
    if (idx >= total) return;
    long long e = idx >> 6;                  // / (HD/4)
    int c = (int)(idx & 63) * 4;
    int s = ei[e];
    int d = ei[(size_t)E + e];
    float w = dinv[s] * dinv[d];
    const float4 v = *(const float4*)(m + (size_t)s * HD + c);
    float* out = agg + (size_t)d * HD + c;
    atomicAdd(out + 0, v.x * w);
    atomicAdd(out + 1, v.y * w);
    atomicAdd(out + 2, v.z * w);
    atomicAdd(out + 3, v.w * w);
}

// h = relu(agg + m * dinv^2 + bias)   (self-loop contribution, conflict-free)
__global__ void selfloop_bias_relu(const float* __restrict__ m,
                                   const float* __restrict__ dinv,
                                   const float* __restrict__ bias,
                                   float* __restrict__ h, int N)
{
    long long idx = (long long)blockIdx.x * blockDim.x + threadIdx.x;
    long long total = (long long)N * HD;
    if (idx >= total) return;
    int c = (int)(idx & (HD - 1));
    long long i = idx >> 8;
    float di = dinv[i];
    h[idx] = fmaxf(h[idx] + m[idx] * di * di + bias[c], 0.0f);
}

// pooling pass 1: each block reduces a row chunk, coalesced across columns
__global__ __launch_bounds__(256) void pool_partial(const float* __restrict__ h,
                                                    float* __restrict__ psum,
                                                    float* __restrict__ pmax,
                                                    int N)
{
    int t = threadIdx.x, b = blockIdx.x, nb = gridDim.x;
    int chunk = (N + nb - 1) / nb;
    int r0 = b * chunk;
    int r1 = min(N, r0 + chunk);
    float s = 0.0f, mx = NEG_HUGE;
    for (int r = r0; r < r1; ++r) {
        float v = h[(size_t)r * HD + t];
        s += v;
        mx = fmaxf(mx, v);
    }
    psum[(size_t)b * HD + t] = s;
    pmax[(size_t)b * HD + t] = mx;
}

// pooling pass 2: g[0:256] = mean, g[256:512] = max
__global__ __launch_bounds__(256) void pool_final(const float* __restrict__ psum,
                                                  const float* __restrict__ pmax,
                                                  float* __restrict__ g,
                                                  int nb, int N)
{
    int t = threadIdx.x;
    float s = 0.0f, mx = NEG_HUGE;
    for (int b = 0; b < nb; ++b) {
        s += psum[(size_t)b * HD + t];
        mx = fmaxf(mx, pmax[(size_t)b * HD + t]);
    }
    g[t] = s / (float)N;
    g[HD + t] = mx;
}

// head MLP: out[EMBD] = relu(g[512] @ w1[512,256] + b1) @ w2[256,128] + b2
__global__ __launch_bounds__(256) void mlp_head(const float* __restrict__ g,
                                                const float* __restrict__ w1,
                                                const float* __restrict__ b1,
                                                const float* __restrict__ w2,
                                                const float* __restrict__ b2,
                                                float* __restrict__ out)
{
    __shared__ float gin[2 * HD];
    __shared__ float hid[HD];
    int t = threadIdx.x;
    gin[t] = g[t];
    gin[HD + t] = g[HD + t];
    __syncthreads();
    float acc = b1[t];
    for (int k = 0; k < 2 * HD; ++k) acc += gin[k] * w1[(size_t)k * HD + t];
    hid[t] = fmaxf(acc, 0.0f);
    __syncthreads();
    if (t < EMBD) {
        float o = b2[t];
        for (int j = 0; j < HD; ++j) o += hid[j] * w2[(size_t)j * EMBD + t];
        out[t] = o;
    }
}

// ---------------------------------------------------------------------------
extern "C" void kernel_launch(void* const* d_in, const int* in_sizes, int n_in,
                              void* d_out, int out_size, void* d_ws, size_t ws_size,
                              hipStream_t stream)
{
    const float* x       = (const float*)d_in[0];
    const int*   ei      = (const int*)d_in[1];   // [2,E] (src row, dst row)
    const float* enc_w1  = (const float*)d_in[2];
    const float* enc_b1  = (const float*)d_in[3];
    const float* enc_w2  = (const float*)d_in[4];
    const float* enc_b2  = (const float*)d_in[5];
    const float* conv_w  = (const float*)d_in[6]; // [3,256,256]
    const float* conv_b  = (const float*)d_in[7]; // [3,256]
    const float* mean_w1 = (const float*)d_in[8];
    const float* mean_b1 = (const float*)d_in[9];
    const float* mean_w2 = (const float*)d_in[10];
    const float* mean_b2 = (const float*)d_in[11];
    const float* lv_w1   = (const float*)d_in[12];
    const float* lv_b1   = (const float*)d_in[13];
    const float* lv_w2   = (const float*)d_in[14];
    const float* lv_b2   = (const float*)d_in[15];

    const int       N = in_sizes[0] / 64;         // 100000 (divisible by 16)
    const long long E = (long long)in_sizes[1] / 2;

    // workspace layout (floats): two ping-pong N x 256 buffers kept hot in the
    // 192MB L2, degree/dinv, pooled vector, pooling partials
    float* B0   = (float*)d_ws;                       // m / enc hidden
    float* B1   = B0 + (size_t)N * HD;                // h / agg (in-place)
    float* DINV = B1 + (size_t)N * HD;
    float* G    = DINV + N;                           // 512 floats
    float* PSUM = G + 2 * HD;                         // 256*256
    float* PMAX = PSUM + 256 * HD;                    // 256*256

    const dim3 gGemm(N / 16);                         // 16-row tiles, 256 cols
    const long long NH = (long long)N * HD;

    // node encoder: B0 = relu(x @ w1 + b1); B1 = B0 @ w2 + b2
    gemm_wmma_f32<<<gGemm, 256, 16 * (64 + 4) * sizeof(float), stream>>>(
        x, enc_w1, enc_b1, B0, N, 64, HD, 1);
    gemm_wmma_f32<<<gGemm, 256, 16 * (HD + 4) * sizeof(float), stream>>>(
        B0, enc_w2, enc_b2, B1, N, HD, HD, 0);

    // degrees -> dinv (in place)
    zero_f32<<<(N + 255) / 256, 256, 0, stream>>>(DINV, N);
    deg_count<<<(unsigned)((E + 255) / 256), 256, 0, stream>>>(ei, DINV, E);
    deg_to_dinv<<<(N + 255) / 256, 256, 0, stream>>>(DINV, N);

    // 3 GCN layers; h lives in B1, m in B0
    for (int l = 0; l < 3; ++l) {
        const float* Wl = conv_w + (size_t)l * HD * HD;
        const float* bl = conv_b + (size_t)l * HD;
        gemm_wmma_f32<<<gGemm, 256, 16 * (HD + 4) * sizeof(float), stream>>>(
            B1, Wl, nullptr, B0, N, HD, HD, 0);               // m = h @ W
        zero_f32<<<(unsigned)((NH + 255) / 256), 256, 0, stream>>>(B1, NH);
        scatter_edges<<<(unsigned)((E * (HD / 4) + 255) / 256), 256, 0, stream>>>(
            ei, B0, DINV, B1, E);
        selfloop_bias_relu<<<(unsigned)((NH + 255) / 256), 256, 0, stream>>>(
            B0, DINV, bl, B1, N);
    }

    // pooling + heads
    pool_partial<<<256, 256, 0, stream>>>(B1, PSUM, PMAX, N);
    pool_final<<<1, 256, 0, stream>>>(PSUM, PMAX, G, 256, N);
    mlp_head<<<1, 256, 0, stream>>>(G, mean_w1, mean_b1, mean_w2, mean_b2,
                                    (float*)d_out);
    mlp_head<<<1, 256, 0, stream>>>(G, lv_w1, lv_b1, lv_w2, lv_b2,
                                    (float*)d_out + EMBD);
}